// GCN_9620726743390
// MI455X (gfx1250) — compile-verified
//
#include <hip/hip_runtime.h>

#define NN 50000
#define NE 800000
#define FN 32
#define FE 16
#define HID 128
#define NL 3

typedef float v2f __attribute__((ext_vector_type(2)));
typedef float v8f __attribute__((ext_vector_type(8)));

// ---------------------------------------------------------------------------
// C[M,128] = maybe_relu(A[M,K] @ W[K,128] + bias1 + bias2)
// One wave32 per 16x16 output tile, V_WMMA_F32_16X16X4_F32, K stepped by 4.
// blockDim = 256 (8 waves); block b handles row-tile b, wave w handles col-tile w.
// M must be a multiple of 16 (50000 and 800000 both are) -> EXEC all ones.
// ---------------------------------------------------------------------------
__global__ __launch_bounds__(256) void gemm_bias_relu(
    const float* __restrict__ A, const float* __restrict__ W,
    const float* __restrict__ bias1, const float* __restrict__ bias2,
    float* __restrict__ C, int M, int K, int do_relu)
{
    const int lane    = threadIdx.x & 31;
    const int colTile = threadIdx.x >> 5;      // 0..7 -> 128 columns
    const int rowTile = blockIdx.x;
    const int r = lane & 15;
    const int g = lane >> 4;

    const int rowA = rowTile * 16 + r;         // A-fragment row for this lane
    const int colN = colTile * 16 + r;         // B/C-fragment column for this lane

    const float* __restrict__ Arow = A + (size_t)rowA * K;

    v8f acc = {};
    for (int k0 = 0; k0 < K; k0 += 4) {
        // A frag: VGPR j = A[r, k0 + 2g + j]  (two consecutive K -> one b64 load)
        v2f a = *(const v2f*)(Arow + k0 + 2 * g);
        // B frag: VGPR j = W[k0 + 2g + j, colN]
        v2f b;
        b.x = W[(size_t)(k0 + 2 * g + 0) * HID + colN];
        b.y = W[(size_t)(k0 + 2 * g + 1) * HID + colN];
        acc = __builtin_amdgcn_wmma_f32_16x16x4_f32(
            /*neg_a=*/false, a, /*neg_b=*/false, b,
            /*c_mod=*/(short)0, acc, /*reuse_a=*/false, /*reuse_b=*/false);
    }

    float bv = 0.0f;
    if (bias1) bv += bias1[colN];
    if (bias2) bv += bias2[colN];

    // C frag: VGPR v -> row (v + 8g)
#pragma unroll
    for (int v = 0; v < 8; ++v) {
        int rowOut = rowTile * 16 + v + 8 * g;
        float val  = acc[v] + bv;
        if (do_relu) val = fmaxf(val, 0.0f);
        C[(size_t)rowOut * HID + colN] = val;
    }
}

// ---------------------------------------------------------------------------
// degree / norm helpers
// ---------------------------------------------------------------------------
__global__ void fill_f32(float* __restrict__ p, float v, int n)
{
    int i = blockIdx.x * blockDim.x + threadIdx.x;
    if (i < n) p[i] = v;
}

__global__ void deg_accum(const int* __restrict__ dst, float* __restrict__ deg, int n)
{
    int i = blockIdx.x * blockDim.x + threadIdx.x;
    if (i < n) atomicAdd(deg + dst[i], 1.0f);
}

__global__ void rsqrt_inplace(float* __restrict__ p, int n)
{
    int i = blockIdx.x * blockDim.x + threadIdx.x;
    if (i < n) {
        float d = p[i];
        p[i] = (d > 0.0f) ? rsqrtf(d) : 0.0f;
    }
}

// ---------------------------------------------------------------------------
// agg[dst] += hw[src] * (dinv[src]*dinv[dst]); one wave32 per edge,
// float4 load + 4x global_atomic_add_f32 per lane (agg fits in L2).
// ---------------------------------------------------------------------------
__global__ __launch_bounds__(256) void scatter_edges(
    const float* __restrict__ hw, const int* __restrict__ src,
    const int* __restrict__ dst, const float* __restrict__ dinv,
    float* __restrict__ agg, int E)
{
    int w    = (blockIdx.x * blockDim.x + threadIdx.x) >> 5;
    int lane = threadIdx.x & 31;
    if (w >= E) return;
    int s = src[w];
    int d = dst[w];
    float n = dinv[s] * dinv[d];
    const float4 v = *(const float4*)(hw + (size_t)s * HID + lane * 4);
    float* ap = agg + (size_t)d * HID + lane * 4;
    atomicAdd(ap + 0, v.x * n);
    atomicAdd(ap + 1, v.y * n);
    atomicAdd(ap + 2, v.z * n);
    atomicAdd(ap + 3, v.w * n);
}

// ---------------------------------------------------------------------------
// h = relu(agg + hw * dinv^2)   (agg already holds h@W_res + b_res + b_conv
//                                plus the scattered edge messages;
//                                hw*dinv^2 is the self-loop message)
// ---------------------------------------------------------------------------
__global__ void combine_relu(const float* __restrict__ agg,
                             const float* __restrict__ hw,
                             const float* __restrict__ dinv,
                             float* __restrict__ h, int total)
{
    int idx = blockIdx.x * blockDim.x + threadIdx.x;
    if (idx >= total) return;
    int node = idx >> 7;
    float di = dinv[node];
    float v  = agg[idx] + hw[idx] * (di * di);
    h[idx]   = fmaxf(v, 0.0f);
}

// ---------------------------------------------------------------------------
// decoder: out[e] = h[row].Wd[0:128] + h[col].Wd[128:256]
//                 + relu(ea[e] @ We + be).Wd[256:384] + bd
// One wave32 per edge; edge encoder recomputed on the fly (avoids a 410MB
// intermediate). Wave reduction via shfl_xor (wave32).
// ---------------------------------------------------------------------------
__global__ __launch_bounds__(256) void decode_edges(
    const float* __restrict__ h,  const float* __restrict__ ea,
    const int* __restrict__ src,  const int* __restrict__ dst,
    const float* __restrict__ We, const float* __restrict__ be,
    const float* __restrict__ Wd, const float* __restrict__ bd,
    float* __restrict__ out, int E)
{
    int e    = (blockIdx.x * blockDim.x + threadIdx.x) >> 5;
    int lane = threadIdx.x & 31;
    if (e >= E) return;

    int r = src[e];
    int c = dst[e];

    // lane k (<16) owns edge_attr[e,k]; broadcast with shfl
    float eav = (lane < FE) ? ea[(size_t)e * FE + lane] : 0.0f;

    float acc = 0.0f;
#pragma unroll
    for (int j = 0; j < 4; ++j) {
        int col = lane + 32 * j;
        acc += h[(size_t)r * HID + col] * Wd[col];
        acc += h[(size_t)c * HID + col] * Wd[HID + col];
        float s = be[col];
#pragma unroll
        for (int k = 0; k < FE; ++k) {
            float a = __shfl(eav, k, 32);
            s += a * We[(size_t)k * HID + col];
        }
        acc += fmaxf(s, 0.0f) * Wd[2 * HID + col];
    }
#pragma unroll
    for (int off = 16; off > 0; off >>= 1)
        acc += __shfl_xor(acc, off, 32);
    if (lane == 0) out[e] = acc + bd[0];
}

// ---------------------------------------------------------------------------
extern "C" void kernel_launch(void* const* d_in, const int* in_sizes, int n_in,
                              void* d_out, int out_size, void* d_ws, size_t ws_size,
                              hipStream_t stream)
{
    const float* x   = (const float*)d_in[0];
    const float* ea  = (const float*)d_in[1];
    const int*   ei  = (const int*)  d_in[2];   // [2, E]: src then dst
    const float* Wn  = (const float*)d_in[3];
    const float* bn  = (const float*)d_in[4];
    const float* We  = (const float*)d_in[5];
    const float* be  = (const float*)d_in[6];
    const float* Wc  = (const float*)d_in[7];   // [3,128,128]
    const float* bc  = (const float*)d_in[8];   // [3,128]
    const float* Wr  = (const float*)d_in[9];
    const float* br  = (const float*)d_in[10];
    const float* Wd  = (const float*)d_in[11];  // [384,1]
    const float* bd  = (const float*)d_in[12];
    float* out = (float*)d_out;

    const int* srcI = ei;
    const int* dstI = ei + NE;

    float* ws   = (float*)d_ws;
    float* h    = ws;                       // [NN,128]
    float* hw   = h   + (size_t)NN * HID;   // [NN,128]
    float* agg  = hw  + (size_t)NN * HID;   // [NN,128]
    float* dinv = agg + (size_t)NN * HID;   // [NN]

    const int nodeElems = NN * HID;

    // 1) node encoder: h = relu(x @ Wn + bn)
    gemm_bias_relu<<<NN / 16, 256, 0, stream>>>(x, Wn, bn, nullptr, h, NN, FN, 1);

    // 2) degrees (self-loop baked in as init=1), then dinv = rsqrt(deg)
    fill_f32<<<(NN + 255) / 256, 256, 0, stream>>>(dinv, 1.0f, NN);
    deg_accum<<<(NE + 255) / 256, 256, 0, stream>>>(dstI, dinv, NE);
    rsqrt_inplace<<<(NN + 255) / 256, 256, 0, stream>>>(dinv, NN);

    // 3) GNN layers
    for (int l = 0; l < NL; ++l) {
        // hw = h @ W_conv[l]
        gemm_bias_relu<<<NN / 16, 256, 0, stream>>>(
            h, Wc + (size_t)l * HID * HID, nullptr, nullptr, hw, NN, HID, 0);
        // agg = h @ W_res + b_res + b_conv[l]   (scatter target, pre-seeded)
        gemm_bias_relu<<<NN / 16, 256, 0, stream>>>(
            h, Wr, br, bc + (size_t)l * HID, agg, NN, HID, 0);
        // agg[dst] += hw[src] * norm
        scatter_edges<<<NE / 8, 256, 0, stream>>>(hw, srcI, dstI, dinv, agg, NE);
        // h = relu(agg + hw * dinv^2)
        combine_relu<<<(nodeElems + 255) / 256, 256, 0, stream>>>(agg, hw, dinv, h, nodeElems);
    }

    // 4) decoder
    decode_edges<<<NE / 8, 256, 0, stream>>>(h, ea, srcI, dstI, We, be, Wd, bd, out, NE);
}